// Retention_7550552506887
// MI455X (gfx1250) — compile-verified
//
#include <hip/hip_runtime.h>
#include <hip/hip_bf16.h>

typedef __attribute__((ext_vector_type(2))) float v2f;
typedef __attribute__((ext_vector_type(8))) float v8f;
typedef unsigned int u32x4 __attribute__((ext_vector_type(4)));
typedef int i32x8 __attribute__((ext_vector_type(8)));
typedef int i32x4 __attribute__((ext_vector_type(4)));

#define B_ 2
#define T_ 4096
#define H_ 16
#define D_ 64
#define C_ 64
#define N_ 64
#define E_ 2080
#define SCALE_ 0.125f
#define SCALE2_ (SCALE_*SCALE_)

// E-split for the inter-chunk scan: 2080 = 26 blocks of 80 rows (5 WMMA M-tiles)
#define EB 80
#define NEB 26
#define ETILES 5

static __device__ __forceinline__ v8f wmma_f32(v2f a, v2f b, v8f c) {
  // D = A(16x4,f32) * B(4x16,f32) + C(16x16,f32)
  return __builtin_amdgcn_wmma_f32_16x16x4_f32(false, a, false, b, (short)0, c,
                                               false, false);
}

static __device__ __forceinline__ void atomAddF(float* p, float v) {
  __hip_atomic_fetch_add(p, v, __ATOMIC_RELAXED, __HIP_MEMORY_SCOPE_AGENT);
}

// TDM: DMA a 2-D f32 tile (tile_d0 x tile_d1) from global (row stride
// stride0 elements) into LDS at lds_addr; optional hardware padding of
// 1 DWORD after every 64 DWORDs reproduces the stride-65 LDS layout.
static __device__ __forceinline__ void tdm_load_2d(unsigned lds_addr,
                                                   const void* gptr,
                                                   unsigned tile_d0,
                                                   unsigned tile_d1,
                                                   unsigned stride0,
                                                   bool pad65) {
  unsigned long long ga = (unsigned long long)(uintptr_t)gptr;
  u32x4 g0;
  g0[0] = 1u;                                   // count=1 (valid user D#)
  g0[1] = lds_addr;                             // LDS byte address
  g0[2] = (unsigned)(ga & 0xFFFFFFFFu);         // global_addr[31:0]
  g0[3] = (unsigned)((ga >> 32) & 0x01FFFFFFu)  // global_addr[56:32]
          | (2u << 30);                         // type=2 ("image")
  unsigned w0 = (2u << 16);                     // data_size = 4 bytes
  if (pad65) w0 |= (1u << 20) | (5u << 22);     // pad_en, every 64 dw, +1 dw
  i32x8 g1;
  g1[0] = (int)w0;
  g1[1] = (int)((tile_d0 & 0xFFFFu) << 16);     // tensor_dim0 = tile_d0 (lo16)
  g1[2] = (int)((tile_d1 & 0xFFFFu) << 16);     // td0 hi16=0 | tensor_dim1 lo16
  g1[3] = (int)((tile_d0 & 0xFFFFu) << 16);     // td1 hi16=0 | tile_dim0
  g1[4] = (int)(tile_d1 & 0xFFFFu);             // tile_dim1 | tile_dim2=0
  g1[5] = (int)stride0;                         // tensor_dim0_stride[31:0]
  g1[6] = 0;                                    // stride0 hi | stride1 lo
  g1[7] = 0;
  i32x4 z4 = {0, 0, 0, 0};                      // groups 2/3 unused (2-D)
  i32x8 z8 = {0, 0, 0, 0, 0, 0, 0, 0};
  __builtin_amdgcn_tensor_load_to_lds(g0, g1, z4, z4, z8, 0);
}

static __device__ __forceinline__ unsigned lds_off(const void* p) {
  return (unsigned)(uintptr_t)p;  // LDS aperture: low 32 bits = LDS offset
}

// ---------------------------------------------------------------- utilities

__global__ void k_zero(float* __restrict__ p, int nelem) {
  int idx = blockIdx.x * blockDim.x + threadIdx.x;
  for (; idx < nelem; idx += gridDim.x * blockDim.x) p[idx] = 0.0f;
}

// per-chunk inclusive cumsum of gate + chunk totals
__global__ void k_gates(const float* __restrict__ gate, float* __restrict__ cgW,
                        float* __restrict__ gsumW) {
  int cid = blockIdx.x * blockDim.x + threadIdx.x;
  if (cid >= B_ * H_ * N_) return;
  int n = cid % N_;
  int h = (cid / N_) % H_;
  int b = cid / (N_ * H_);
  float cum = 0.0f;
  for (int c = 0; c < C_; ++c) {
    cum += gate[(b * T_ + n * C_ + c) * H_ + h];
    cgW[(b * H_ + h) * T_ + n * C_ + c] = cum;
  }
  gsumW[(b * H_ + h) * N_ + n] = cum;
}

// triu index tables for the symmetric degree-2 feature map
__global__ void k_triu(int* __restrict__ eI, int* __restrict__ eJ,
                       float* __restrict__ ecoef) {
  int e = blockIdx.x * blockDim.x + threadIdx.x;
  if (e >= E_) return;
  int i = 0, rem = e, len = D_;
  while (rem >= len) { rem -= len; len--; i++; }
  eI[e] = i;
  eJ[e] = i + rem;
  ecoef[e] = (rem == 0) ? 1.0f : 1.41421356237309515f;
}

// --------------------------------------------------- intra-chunk attention
// one block per (b,h,n): qk^2 * decay * mask, num = s@v, den = rowsum(s)

__global__ __launch_bounds__(128) void k_intra(
    const float* __restrict__ q, const float* __restrict__ k,
    const float* __restrict__ v, const float* __restrict__ cgW,
    float* __restrict__ out, float* __restrict__ denW) {
  __shared__ float q_s[64 * 65];  // q tile; reused as masked score matrix s
  __shared__ float k_s[64 * 65];
  __shared__ float v_s[64 * 65];
  __shared__ float cg_s[64];

  const int tid = threadIdx.x;
  const int wave = tid >> 5, lane = tid & 31;
  const int hl = lane >> 4, ln = lane & 15;

  const int bid = blockIdx.x;
  const int n = bid & 63;
  const int h = (bid >> 6) & 15;
  const int b = bid >> 10;

  const int t0 = n * C_;
  const int gbase = ((b * T_ + t0) * H_ + h) * D_;  // (c,d) stride = H_*D_, 1

  // TDM: DMA the three 64x64 tiles straight into LDS (stride-65 padded)
  if (wave == 0) {
    tdm_load_2d(lds_off(q_s), q + gbase, 64, 64, H_ * D_, true);
    tdm_load_2d(lds_off(k_s), k + gbase, 64, 64, H_ * D_, true);
    tdm_load_2d(lds_off(v_s), v + gbase, 64, 64, H_ * D_, true);
    __builtin_amdgcn_s_wait_tensorcnt(0);
  }
  if (tid < 64) cg_s[tid] = cgW[(b * H_ + h) * T_ + t0 + tid];
  __syncthreads();

  const int ti = wave;  // each wave owns rows [16*ti, 16*ti+16)

  // qk = q @ k^T over D=64 (16 WMMA k-steps), all 4 column tiles
  v8f acc[4];
  for (int tj = 0; tj < 4; ++tj) {
    v8f a8 = {};
    for (int kb = 0; kb < 64; kb += 4) {
      int kk = kb + 2 * hl;
      v2f af, bf;
      af.x = q_s[(ti * 16 + ln) * 65 + kk];
      af.y = q_s[(ti * 16 + ln) * 65 + kk + 1];
      bf.x = k_s[(tj * 16 + ln) * 65 + kk];      // B[d][m] = k[m][d]
      bf.y = k_s[(tj * 16 + ln) * 65 + kk + 1];
      a8 = wmma_f32(af, bf, a8);
    }
    acc[tj] = a8;
  }
  // elementwise: (qk*scale)^2 * exp(cg[c]-cg[m]) * causal; write s over q_s
  for (int tj = 0; tj < 4; ++tj) {
    for (int r = 0; r < 8; ++r) {
      int crow = ti * 16 + r + 8 * hl;
      int mcol = tj * 16 + ln;
      float x = acc[tj][r] * SCALE_;
      x = x * x;
      x = (mcol <= crow) ? x * __expf(cg_s[crow] - cg_s[mcol]) : 0.0f;
      q_s[crow * 65 + mcol] = x;
    }
  }
  __syncthreads();

  // den = rowsum(s)
  if (tid < 64) {
    float s = 0.0f;
    for (int m = 0; m < 64; ++m) s += q_s[tid * 65 + m];
    denW[(b * H_ + h) * T_ + t0 + tid] = s;
  }

  // num = s @ v over C=64, plain store (inter kernel atomically adds later)
  for (int tj = 0; tj < 4; ++tj) {
    v8f a8 = {};
    for (int kb = 0; kb < 64; kb += 4) {
      int kk = kb + 2 * hl;
      v2f af, bf;
      af.x = q_s[(ti * 16 + ln) * 65 + kk];
      af.y = q_s[(ti * 16 + ln) * 65 + kk + 1];
      bf.x = v_s[kk * 65 + tj * 16 + ln];        // B[m][d] = v[m][d]
      bf.y = v_s[(kk + 1) * 65 + tj * 16 + ln];
      a8 = wmma_f32(af, bf, a8);
    }
    for (int r = 0; r < 8; ++r) {
      int crow = ti * 16 + r + 8 * hl;
      int dcol = tj * 16 + ln;
      out[gbase + crow * (H_ * D_) + dcol] = a8[r];
    }
  }
}

// -------------------------------------------- inter-chunk scan, E-split
// one block per (b,h,eblock): sequential scan over 64 chunks; state slice
// S[e0:e0+80, 0:64] and sk[e0:e0+80] live in LDS for the whole scan.

__global__ __launch_bounds__(128) void k_inter(
    const float* __restrict__ q, const float* __restrict__ k,
    const float* __restrict__ v, const float* __restrict__ S0,
    const float* __restrict__ sk0, const float* __restrict__ cgW,
    const float* __restrict__ gsumW, const int* __restrict__ eI,
    const int* __restrict__ eJ, const float* __restrict__ ecoef,
    float* __restrict__ out, float* __restrict__ denW) {
  __shared__ float S_s[EB * 65];   // state slice, stride 65
  __shared__ float P_s[64 * 81];   // pq then pk, stride 81
  __shared__ float v_s[64 * 65];
  __shared__ float cg_s[64], qdec_s[64], pkf_s[64];
  __shared__ float sk_s[EB];

  const int tid = threadIdx.x;
  const int wave = tid >> 5, lane = tid & 31;
  const int hl = lane >> 4, ln = lane & 15;

  const int bid = blockIdx.x;
  const int eb = bid % NEB;
  const int h = (bid / NEB) % H_;
  const int b = bid / (NEB * H_);
  const int e0 = eb * EB;

  // TDM: state slice 80 rows x 64 cols (contiguous rows, pad to stride 65)
  const int sbase = ((b * H_ + h) * E_ + e0) * D_;
  if (wave == 0) {
    tdm_load_2d(lds_off(S_s), S0 + sbase, 64, EB, 64, true);
    __builtin_amdgcn_s_wait_tensorcnt(0);
  }
  for (int idx = tid; idx < EB; idx += 128)
    sk_s[idx] = sk0[(b * H_ + h) * E_ + e0 + idx];
  __syncthreads();

  for (int n = 0; n < N_; ++n) {
    const int t0 = n * C_;
    const int gbase = ((b * T_ + t0) * H_ + h) * D_;
    const float gs = gsumW[(b * H_ + h) * N_ + n];
    const float egs = __expf(gs);

    // TDM: v tile for this chunk
    if (wave == 0) {
      tdm_load_2d(lds_off(v_s), v + gbase, 64, 64, H_ * D_, true);
    }
    if (tid < 64) {
      float cgv = cgW[(b * H_ + h) * T_ + t0 + tid];
      cg_s[tid] = cgv;
      qdec_s[tid] = __expf(cgv);
      pkf_s[tid] = __expf(gs - cgv);
    }
    if (tid < 64 && n + 1 < N_) {  // prefetch next chunk q/k (phi building)
      const float* qn = q + gbase + (64 + tid) * (H_ * D_);
      const float* kn = k + gbase + (64 + tid) * (H_ * D_);
      __builtin_prefetch(qn, 0, 1);      __builtin_prefetch(qn + 32, 0, 1);
      __builtin_prefetch(kn, 0, 1);      __builtin_prefetch(kn + 32, 0, 1);
    }
    if (wave == 0) __builtin_amdgcn_s_wait_tensorcnt(0);
    __syncthreads();

    // build pq slice: phi(q)*SCALE^2
    for (int idx = tid; idx < 64 * EB; idx += 128) {
      int c = idx / EB, el = idx % EB;
      int e = e0 + el;
      const float* qr = q + gbase + c * (H_ * D_);
      P_s[c * 81 + el] = qr[eI[e]] * qr[eJ[e]] * ecoef[e] * SCALE2_;
    }
    __syncthreads();

    // num += qdec * (pq @ S_slice)   [64 x EB] @ [EB x 64]
    for (int tj = 0; tj < 4; ++tj) {
      v8f a8 = {};
      for (int kb = 0; kb < EB; kb += 4) {
        int kk = kb + 2 * hl;
        v2f af, bf;
        af.x = P_s[(wave * 16 + ln) * 81 + kk];
        af.y = P_s[(wave * 16 + ln) * 81 + kk + 1];
        bf.x = S_s[kk * 65 + tj * 16 + ln];
        bf.y = S_s[(kk + 1) * 65 + tj * 16 + ln];
        a8 = wmma_f32(af, bf, a8);
      }
      for (int r = 0; r < 8; ++r) {
        int crow = wave * 16 + r + 8 * hl;
        int dcol = tj * 16 + ln;
        atomAddF(&out[gbase + crow * (H_ * D_) + dcol], a8[r] * qdec_s[crow]);
      }
    }
    // den += qdec * (pq . sk)
    {
      int c = tid & 63;
      int hf = tid >> 6;
      float s = 0.0f;
      for (int el = hf * (EB / 2); el < (hf + 1) * (EB / 2); ++el)
        s += P_s[c * 81 + el] * sk_s[el];
      atomAddF(&denW[(b * H_ + h) * T_ + t0 + c], s * qdec_s[c]);
    }
    __syncthreads();

    // build pk slice (overwrites P_s): phi(k) * exp(gs - cg[c])
    for (int idx = tid; idx < 64 * EB; idx += 128) {
      int c = idx / EB, el = idx % EB;
      int e = e0 + el;
      const float* kr = k + gbase + c * (H_ * D_);
      P_s[c * 81 + el] = kr[eI[e]] * kr[eJ[e]] * ecoef[e] * pkf_s[c];
    }
    __syncthreads();

    // sk = egs*sk + colsum(pk)
    for (int el = tid; el < EB; el += 128) {
      float s = 0.0f;
      for (int c = 0; c < 64; ++c) s += P_s[c * 81 + el];
      sk_s[el] = egs * sk_s[el] + s;
    }
    // S = egs*S + pk^T @ v   [EB x 64] += [EB x 64(c)] @ [64(c) x 64]
    for (int et = wave; et < ETILES; et += 4) {
      for (int tj = 0; tj < 4; ++tj) {
        v8f a8;
        for (int r = 0; r < 8; ++r)
          a8[r] = egs * S_s[(et * 16 + r + 8 * hl) * 65 + tj * 16 + ln];
        for (int kb = 0; kb < 64; kb += 4) {
          int kk = kb + 2 * hl;
          v2f af, bf;
          af.x = P_s[kk * 81 + et * 16 + ln];        // A[e][c] = pk[c][e]
          af.y = P_s[(kk + 1) * 81 + et * 16 + ln];
          bf.x = v_s[kk * 65 + tj * 16 + ln];
          bf.y = v_s[(kk + 1) * 65 + tj * 16 + ln];
          a8 = wmma_f32(af, bf, a8);
        }
        for (int r = 0; r < 8; ++r)
          S_s[(et * 16 + r + 8 * hl) * 65 + tj * 16 + ln] = a8[r];
      }
    }
    __syncthreads();
  }
}

// ---------------------------------------------------------------- finalize

__global__ void k_final(float* __restrict__ out, const float* __restrict__ denW) {
  int idx = blockIdx.x * blockDim.x + threadIdx.x;
  const int total = B_ * T_ * H_ * D_;
  for (; idx < total; idx += gridDim.x * blockDim.x) {
    int h = (idx >> 6) & 15;
    int t = (idx >> 10) & 4095;
    int b = idx >> 22;
    float dn = denW[(b * H_ + h) * T_ + t];
    out[idx] = out[idx] / fmaxf(dn, 1.0f);
  }
}

// ---------------------------------------------------------------- launcher

extern "C" void kernel_launch(void* const* d_in, const int* in_sizes, int n_in,
                              void* d_out, int out_size, void* d_ws,
                              size_t ws_size, hipStream_t stream) {
  (void)in_sizes; (void)n_in; (void)out_size; (void)ws_size;
  const float* q   = (const float*)d_in[0];
  const float* k   = (const float*)d_in[1];
  const float* v   = (const float*)d_in[2];
  const float* g   = (const float*)d_in[3];
  const float* S0  = (const float*)d_in[4];
  const float* sk0 = (const float*)d_in[5];
  float* out = (float*)d_out;

  float* wsf   = (float*)d_ws;            // ~1.06 MB of workspace used
  float* cgW   = wsf;                     // [B,H,T]
  float* gsumW = cgW + B_ * H_ * T_;      // [B,H,N]
  float* denW  = gsumW + B_ * H_ * N_;    // [B,H,T]
  int*   eI    = (int*)(denW + B_ * H_ * T_);
  int*   eJ    = eI + E_;
  float* ecoef = (float*)(eJ + E_);

  k_zero<<<1024, 256, 0, stream>>>(out, B_ * T_ * H_ * D_);
  k_zero<<<64, 256, 0, stream>>>(denW, B_ * H_ * T_);
  k_gates<<<(B_ * H_ * N_ + 255) / 256, 256, 0, stream>>>(g, cgW, gsumW);
  k_triu<<<(E_ + 255) / 256, 256, 0, stream>>>(eI, eJ, ecoef);
  k_intra<<<B_ * H_ * N_, 128, 0, stream>>>(q, k, v, cgW, out, denW);
  k_inter<<<B_ * H_ * NEB, 128, 0, stream>>>(q, k, v, S0, sk0, cgW, gsumW, eI,
                                             eJ, ecoef, out, denW);
  k_final<<<2048, 256, 0, stream>>>(out, denW);
}